// GraphMambaLayer_69234872812073
// MI455X (gfx1250) — compile-verified
//
#include <hip/hip_runtime.h>
#include <hip/hip_bf16.h>

typedef __attribute__((ext_vector_type(2))) float v2f;
typedef __attribute__((ext_vector_type(8))) float v8f;
typedef __attribute__((ext_vector_type(4))) unsigned int v4u;
typedef __attribute__((ext_vector_type(4))) int v4i;
typedef __attribute__((ext_vector_type(8))) int v8i;

#define NNODE 50000
#define DIM   128
#define NEDGE 500000
#define BB    4
#define LL    4096
#define TT    (BB * LL)
#define DSTATE 16
#define DTRANK 8

#define USE_TDM 1

#if USE_TDM
// Build a D# and issue TENSOR_LOAD_TO_LDS for a 2D tile of a row-major f32
// tensor: tile is tile_rows x row_elems, starting at gaddr (byte address of
// tile origin), source row stride = src_stride elems, tensor is tens_rows
// rows of tens_cols elems (OOB rows/cols read as zero). LDS destination rows
// are padded by pad_dw dwords every row_elems dwords (row_elems must be a
// power of two >= 2; pad via TDM pad_interval/pad_amount).
__device__ __forceinline__ void tdm_load_tile_f32(
    unsigned lds_byte_addr, const float* gaddr, unsigned row_elems,
    unsigned tile_rows, unsigned tens_cols, unsigned tens_rows,
    unsigned src_stride, unsigned pad_dw) {
  // pad_interval code: 1<<code 8-byte units before pad => code = log2(row_elems/2)
  unsigned padi = 31u - (unsigned)__clz((int)(row_elems >> 1));
  unsigned pada = pad_dw - 1u;  // code = dwords-1
  unsigned long long ga = (unsigned long long)(const void*)gaddr;
  v4u g0;
  g0[0] = 1u;                                   // count=1, user descriptor
  g0[1] = lds_byte_addr;                        // lds_addr
  g0[2] = (unsigned)(ga & 0xFFFFFFFFu);         // global_addr[31:0]
  g0[3] = (unsigned)((ga >> 32) & 0x01FFFFFFu) | (2u << 30);  // addr hi | type=2
  v8i g1;
  g1[0] = (int)((2u << 16) | (1u << 20) | (padi << 22) | (pada << 25));
  g1[1] = (int)((tens_cols & 0xFFFFu) << 16);                 // tensor_dim0 lo
  g1[2] = (int)(((tens_cols >> 16) & 0xFFFFu) |
                ((tens_rows & 0xFFFFu) << 16));               // td0 hi | td1 lo
  g1[3] = (int)(((tens_rows >> 16) & 0xFFFFu) |
                ((row_elems & 0xFFFFu) << 16));               // td1 hi | tile_dim0
  g1[4] = (int)(tile_rows & 0xFFFFu);                         // tile_dim1 (dim2=0)
  g1[5] = (int)src_stride;                                    // dim0_stride[31:0]
  g1[6] = 0;                                                  // stride hi, stride1 lo
  g1[7] = 0;
  v4i z4 = {0, 0, 0, 0};
  v8i z8 = {0, 0, 0, 0, 0, 0, 0, 0};
  __builtin_amdgcn_tensor_load_to_lds(g0, g1, z4, z4, z8, 0);
}
#endif

// ---------------------------------------------------------------------------
// Generic WMMA f32 GEMM: C[M,Nc] = A[M,K] @ W[K,Nc] (+bias, +ReLU, +resid)
// 256 threads (8 waves); block computes a 128x128 tile. TDM stages both the
// A tile (padded row stride K+4) and W tile (padded row stride 132) into LDS;
// wave w owns 16 rows, sweeps 8 column subtiles with V_WMMA_F32_16X16X4_F32.
// ---------------------------------------------------------------------------
template <int K, bool BIAS, bool RELU, bool RES>
__global__ __launch_bounds__(256) void gemm_wmma(
    const float* __restrict__ A, const float* __restrict__ W,
    const float* __restrict__ bias, const float* __restrict__ resid,
    float* __restrict__ C, int M, int Nc) {
  constexpr int SA = K + 4;   // padded LDS row stride for A tile (bank spread)
  constexpr int SW = 132;     // padded LDS row stride for W tile
  extern __shared__ float smem[];
  float* As = smem;                 // [128][SA]
  float* Ws = smem + 128 * SA;      // [K][SW]

  const int tid = threadIdx.x;
  const int m0 = blockIdx.x * 128;
  const int nb0 = blockIdx.y * 128;

#if USE_TDM
  // Wave 0 issues both tile DMAs; scalarized branch so waves 1-7 skip issue
  // (TDM ignores EXEC, so exec-masking alone would duplicate the DMA).
  if (__builtin_amdgcn_readfirstlane(tid >> 5) == 0) {
    unsigned lds_a = (unsigned)(unsigned long long)(void*)As;
    unsigned lds_w = (unsigned)(unsigned long long)(void*)Ws;
    // A tile: 128 rows x K elems from [M,K]; OOB rows (>=M) read as zero.
    tdm_load_tile_f32(lds_a, A + (size_t)m0 * K, K, 128, K, (unsigned)M, K, 4);
    // W tile: K rows x 128 elems from [K,Nc] starting at column nb0.
    tdm_load_tile_f32(lds_w, W + nb0, 128, K, (unsigned)(Nc - nb0), K, Nc, 4);
    __builtin_amdgcn_s_wait_tensorcnt(0);
  }
#else
  constexpr int AV = 128 * (K / 4);
  for (int i = tid; i < AV; i += 256) {
    int r = i / (K / 4);
    int c4 = (i % (K / 4)) * 4;
    int gr = m0 + r;
    float4 v = make_float4(0.f, 0.f, 0.f, 0.f);
    if (gr < M) v = *reinterpret_cast<const float4*>(A + (size_t)gr * K + c4);
    *reinterpret_cast<float4*>(As + r * SA + c4) = v;
  }
  constexpr int WV = K * 32;
  for (int i = tid; i < WV; i += 256) {
    int k = i / 32;
    int c4 = (i % 32) * 4;
    float4 v = *reinterpret_cast<const float4*>(W + (size_t)k * Nc + nb0 + c4);
    *reinterpret_cast<float4*>(Ws + k * SW + c4) = v;
  }
#endif
  __syncthreads();

  const int wv = tid >> 5;          // wave id 0..7 -> row subtile
  const int lane = tid & 31;
  const int lh = lane & 15;
  const int hi = (lane >> 4) & 1;   // 0: K pair {0,1}, 1: K pair {2,3}
  const int koff = hi * 2;

  // Preload all per-lane A operand pairs once (ds_load_b64 each).
  constexpr int KS = K / 4;
  const float* Arow = As + (wv * 16 + lh) * SA;
  v2f areg[KS];
#pragma unroll
  for (int m = 0; m < KS; ++m) {
    areg[m][0] = Arow[4 * m + koff];
    areg[m][1] = Arow[4 * m + koff + 1];
  }

  for (int ct = 0; ct < 8; ++ct) {
    const int ncol = ct * 16 + lh;
    const float* Bcol = Ws + (size_t)koff * SW + ncol;
    v8f acc = {0.f, 0.f, 0.f, 0.f, 0.f, 0.f, 0.f, 0.f};
    // Software-pipelined B fetch: keep one DS load in flight past each WMMA.
    v2f bcur, bnext;
    bcur[0] = Bcol[0];
    bcur[1] = Bcol[SW];
#pragma unroll
    for (int m = 0; m < KS; ++m) {
      if (m + 1 < KS) {
        bnext[0] = Bcol[(4 * (m + 1)) * SW];
        bnext[1] = Bcol[(4 * (m + 1) + 1) * SW];
      } else {
        bnext = bcur;
      }
      acc = __builtin_amdgcn_wmma_f32_16x16x4_f32(
          false, areg[m], false, bcur, (short)0, acc, false, false);
      bcur = bnext;
    }
    const int gcol = nb0 + ct * 16 + lh;
    float bv = BIAS ? bias[gcol] : 0.f;
#pragma unroll
    for (int v = 0; v < 8; ++v) {
      int gr = m0 + wv * 16 + v + hi * 8;
      if (gr < M) {
        float val = acc[v] + bv;
        if (RELU) val = val > 0.f ? val : 0.f;
        if (RES) val += resid[(size_t)gr * Nc + gcol];
        C[(size_t)gr * Nc + gcol] = val;
      }
    }
  }
}

// ---------------------------------------------------------------------------
// Edge aggregation: sigma = sigmoid(Dx[dst]+Ex[src]); atomic num/den scatter.
// ---------------------------------------------------------------------------
__global__ __launch_bounds__(256) void edge_kernel(
    const int* __restrict__ edges, const float* __restrict__ Bx,
    const float* __restrict__ Dx, const float* __restrict__ Ex,
    float* __restrict__ num, float* __restrict__ den, int E) {
  int gid = blockIdx.x * 256 + threadIdx.x;
  int e = gid >> 2;
  if (e >= E) return;
  int q = (gid & 3) * 32;
  int src = edges[e];
  int dst = edges[E + e];
  const float* dp = Dx + (size_t)dst * DIM + q;
  const float* ep = Ex + (size_t)src * DIM + q;
  const float* bp = Bx + (size_t)src * DIM + q;
  float* np = num + (size_t)dst * DIM + q;
  float* de = den + (size_t)dst * DIM + q;
#pragma unroll 8
  for (int c = 0; c < 32; ++c) {
    float s = 1.f / (1.f + __expf(-(dp[c] + ep[c])));
    atomicAdd(np + c, s * bp[c]);
    atomicAdd(de + c, s);
  }
}

// h_local = x + relu(Ax + num/(den+1e-6))
__global__ void node_combine(const float* __restrict__ x,
                             const float* __restrict__ Ax,
                             const float* __restrict__ num,
                             const float* __restrict__ den,
                             float* __restrict__ hl, int n) {
  int i = blockIdx.x * 256 + threadIdx.x;
  if (i < n) {
    float o = Ax[i] + num[i] / (den[i] + 1e-6f);
    o = o > 0.f ? o : 0.f;
    hl[i] = x[i] + o;
  }
}

// x_seq[t,:] = emb[seq[t],:]
__global__ void gather_rows(const float* __restrict__ emb,
                            const int* __restrict__ seq,
                            float* __restrict__ out, int T) {
  int gid = blockIdx.x * 256 + threadIdx.x;
  int t = gid >> 5;
  if (t >= T) return;
  int c = (gid & 31) * 4;
  int r = seq[t];
  *reinterpret_cast<float4*>(out + (size_t)t * DIM + c) =
      *reinterpret_cast<const float4*>(emb + (size_t)r * DIM + c);
}

// depthwise causal conv(K=4) + bias + SiLU; input xm = xz[:, 0:128]
__global__ void conv_silu(const float* __restrict__ xz,
                          const float* __restrict__ cw,
                          const float* __restrict__ cb,
                          float* __restrict__ xm2, int T) {
  int gid = blockIdx.x * 256 + threadIdx.x;
  if (gid >= T * DIM) return;
  int d = gid & (DIM - 1);
  int t = gid >> 7;
  int l = t & (LL - 1);
  float acc = cb[d];
#pragma unroll
  for (int k = 0; k < 4; ++k) {
    int ll = l - 3 + k;
    if (ll >= 0) acc += cw[d * 4 + k] * xz[(size_t)(t - 3 + k) * 256 + d];
  }
  xm2[(size_t)t * DIM + d] = acc / (1.f + __expf(-acc));
}

// x_dbl[t, 0:40] = xm2[t,:] @ W_x
__global__ void xdbl_kernel(const float* __restrict__ xm2,
                            const float* __restrict__ Wx,
                            float* __restrict__ xd, int T) {
  int gid = blockIdx.x * 256 + threadIdx.x;
  if (gid >= T * 40) return;
  int t = gid / 40, j = gid % 40;
  const float* xr = xm2 + (size_t)t * DIM;
  float s = 0.f;
#pragma unroll 8
  for (int k = 0; k < DIM; ++k) s += xr[k] * Wx[k * 40 + j];
  xd[(size_t)t * 40 + j] = s;
}

// Selective scan: one block per batch, one thread per channel d.
__global__ __launch_bounds__(128) void scan_kernel(
    const float* __restrict__ xd, const float* __restrict__ xm2,
    const float* __restrict__ xz, const float* __restrict__ Wdt,
    const float* __restrict__ bdt, const float* __restrict__ Alog,
    const float* __restrict__ Dssm, float* __restrict__ y) {
  __shared__ float sh[40];
  int b = blockIdx.x;
  int d = threadIdx.x;
  float Ar[DSTATE];
#pragma unroll
  for (int n = 0; n < DSTATE; ++n) Ar[n] = -__expf(Alog[d * DSTATE + n]);
  float wdt[DTRANK];
#pragma unroll
  for (int r = 0; r < DTRANK; ++r) wdt[r] = Wdt[r * DIM + d];
  float bd = bdt[d], Dd = Dssm[d];
  float h[DSTATE];
#pragma unroll
  for (int n = 0; n < DSTATE; ++n) h[n] = 0.f;
  for (int l = 0; l < LL; ++l) {
    int t = b * LL + l;
    if (d < 40) sh[d] = xd[(size_t)t * 40 + d];
    __syncthreads();
    float dtv = bd;
#pragma unroll
    for (int r = 0; r < DTRANK; ++r) dtv += sh[r] * wdt[r];
    float delta = dtv > 20.f ? dtv : __logf(1.f + __expf(dtv));  // softplus
    float xv = xm2[(size_t)t * DIM + d];
    float acc = 0.f;
#pragma unroll
    for (int n = 0; n < DSTATE; ++n) {
      float dA = __expf(delta * Ar[n]);
      h[n] = dA * h[n] + delta * sh[8 + n] * xv;
      acc += h[n] * sh[24 + n];
    }
    float zv = xz[(size_t)t * 256 + 128 + d];
    float silz = zv / (1.f + __expf(-zv));
    y[(size_t)t * DIM + d] = (acc + xv * Dd) * silz;
    __syncthreads();
  }
}

// h = mo + h_local[seq[t]]
__global__ void combine_h(const float* __restrict__ mo,
                          const float* __restrict__ hl,
                          const int* __restrict__ seq,
                          float* __restrict__ h, int T) {
  int gid = blockIdx.x * 256 + threadIdx.x;
  if (gid >= T * DIM) return;
  int t = gid >> 7, d = gid & (DIM - 1);
  h[gid] = mo[gid] + hl[(size_t)seq[t] * DIM + d];
}

// out[seq[t],:] = 0.5*emb[seq[t],:] + 0.5*h_final[t,:]
__global__ void scatter_out(const float* __restrict__ emb,
                            const int* __restrict__ seq,
                            const float* __restrict__ hf,
                            float* __restrict__ out, int T) {
  int gid = blockIdx.x * 256 + threadIdx.x;
  if (gid >= T * DIM) return;
  int t = gid >> 7, d = gid & (DIM - 1);
  size_t r = (size_t)seq[t] * DIM + d;
  out[r] = 0.5f * emb[r] + 0.5f * hf[gid];
}

// ---------------------------------------------------------------------------
extern "C" void kernel_launch(void* const* d_in, const int* in_sizes, int n_in,
                              void* d_out, int out_size, void* d_ws,
                              size_t ws_size, hipStream_t stream) {
  const float* emb    = (const float*)d_in[0];
  const int*   edges  = (const int*)d_in[1];
  const int*   seq    = (const int*)d_in[2];
  const float* WA = (const float*)d_in[3];  const float* bA = (const float*)d_in[4];
  const float* WB = (const float*)d_in[5];  const float* bBv = (const float*)d_in[6];
  const float* WD = (const float*)d_in[7];  const float* bDv = (const float*)d_in[8];
  const float* WE = (const float*)d_in[9];  const float* bEv = (const float*)d_in[10];
  const float* W_in  = (const float*)d_in[11];
  const float* convw = (const float*)d_in[12];
  const float* convb = (const float*)d_in[13];
  const float* W_x   = (const float*)d_in[14];
  const float* W_dt  = (const float*)d_in[15];
  const float* b_dt  = (const float*)d_in[16];
  const float* A_log = (const float*)d_in[17];
  const float* D_ssm = (const float*)d_in[18];
  const float* W_out = (const float*)d_in[19];
  const float* W1 = (const float*)d_in[20]; const float* b1 = (const float*)d_in[21];
  const float* W2 = (const float*)d_in[22]; const float* b2 = (const float*)d_in[23];

  const size_t ND = (size_t)NNODE * DIM;
  const size_t BLD = (size_t)TT * DIM;
  float* ws = (float*)d_ws;
  // GCN scratch
  float* gA = ws;              // [N,D]
  float* gB = ws + ND;
  float* gD = ws + 2 * ND;
  float* gE = ws + 3 * ND;
  float* num = ws + 4 * ND;    // [N,D]
  float* den = ws + 5 * ND;    // [N,D]
  float* hloc = ws + 6 * ND;   // [N,D]
  // Mamba scratch reuses [0, 4*ND) after GCN phase
  float* q = ws;
  float* xseq = q;             // [T,128]
  float* xz   = q + BLD;       // [T,256]
  float* xm2  = q + 3 * BLD;   // [T,128]
  float* xd   = q + 4 * BLD;   // [T,40]
  float* ybuf = q + 5 * BLD;   // [T,128]
  float* hbuf = q + 6 * BLD;   // [T,128]
  float* tbuf = q + 7 * BLD;   // [T,256]
  float* mo = xz;              // out_proj result reuses xz region
  float* hfinal = ybuf;        // final h reuses ybuf region

  const dim3 blk(256);
  const size_t smem128 = (size_t)(128 * (128 + 4) + 128 * 132) * sizeof(float);
  const size_t smem256 = (size_t)(128 * (256 + 4) + 256 * 132) * sizeof(float);

  // ---- GatedGCN ----
  (void)hipMemsetAsync(num, 0, 2 * ND * sizeof(float), stream);  // num + den
  const int gxN = (NNODE + 127) / 128;
  gemm_wmma<128, true, false, false><<<dim3(gxN, 1), blk, smem128, stream>>>(
      emb, WA, bA, nullptr, gA, NNODE, DIM);
  gemm_wmma<128, true, false, false><<<dim3(gxN, 1), blk, smem128, stream>>>(
      emb, WB, bBv, nullptr, gB, NNODE, DIM);
  gemm_wmma<128, true, false, false><<<dim3(gxN, 1), blk, smem128, stream>>>(
      emb, WD, bDv, nullptr, gD, NNODE, DIM);
  gemm_wmma<128, true, false, false><<<dim3(gxN, 1), blk, smem128, stream>>>(
      emb, WE, bEv, nullptr, gE, NNODE, DIM);
  edge_kernel<<<(NEDGE * 4 + 255) / 256, blk, 0, stream>>>(edges, gB, gD, gE,
                                                           num, den, NEDGE);
  node_combine<<<((int)ND + 255) / 256, blk, 0, stream>>>(emb, gA, num, den,
                                                          hloc, (int)ND);

  // ---- Mamba ----
  gather_rows<<<(TT * 32 + 255) / 256, blk, 0, stream>>>(emb, seq, xseq, TT);
  gemm_wmma<128, false, false, false><<<dim3(TT / 128, 2), blk, smem128, stream>>>(
      xseq, W_in, nullptr, nullptr, xz, TT, 256);
  conv_silu<<<(TT * DIM + 255) / 256, blk, 0, stream>>>(xz, convw, convb, xm2, TT);
  xdbl_kernel<<<(TT * 40 + 255) / 256, blk, 0, stream>>>(xm2, W_x, xd, TT);
  scan_kernel<<<BB, 128, 0, stream>>>(xd, xm2, xz, W_dt, b_dt, A_log, D_ssm, ybuf);
  gemm_wmma<128, false, false, true><<<dim3(TT / 128, 1), blk, smem128, stream>>>(
      ybuf, W_out, nullptr, xseq, mo, TT, DIM);  // + h_in1 residual
  combine_h<<<(TT * DIM + 255) / 256, blk, 0, stream>>>(mo, hloc, seq, hbuf, TT);

  // ---- FFN ----
  gemm_wmma<128, true, true, false><<<dim3(TT / 128, 2), blk, smem128, stream>>>(
      hbuf, W1, b1, nullptr, tbuf, TT, 256);
  gemm_wmma<256, true, false, true><<<dim3(TT / 128, 1), blk, smem256, stream>>>(
      tbuf, W2, b2, hbuf, hfinal, TT, DIM);

  // ---- Output scatter ----
  (void)hipMemcpyAsync(d_out, (const void*)emb, ND * sizeof(float),
                       hipMemcpyDeviceToDevice, stream);
  scatter_out<<<(TT * DIM + 255) / 256, blk, 0, stream>>>(emb, seq, hfinal,
                                                          (float*)d_out, TT);
}